// YoloLoss_56642028700231
// MI455X (gfx1250) — compile-verified
//
#include <hip/hip_runtime.h>
#include <hip/hip_bf16.h>

// YOLO loss, MI455X (gfx1250, wave32).
// Memory-bound: ~192.7 MB streamed -> ~8.3 us floor @ 23.3 TB/s.
// Strategy: float2 (b64) streaming loads, per-thread cell loss, WMMA f32
// 16x16x4 wave reduction (full fp32), deterministic 2-stage block reduce.

#define S_GRID 14
#define CH 30
#define EPSF 1e-20f

typedef float v2f __attribute__((ext_vector_type(2)));
typedef float v8f __attribute__((ext_vector_type(8)));

// ---- wave32 reduction: V_WMMA_F32_16X16X4_F32 with B = ones -----------------
// A (16x4 f32): lane l holds {partial, 0} at (M = l%16, K = 2*(l/16) .. +1).
// D[m][n] = sum_k A[m][k] = partial(m) + partial(m+16). Per-lane sum of the 8
// D VGPRs covers M=0..7 (lanes 0-15) or M=8..15 (lanes 16-31); one xor-16
// shuffle completes the 32-lane sum. EXEC is all-1s at every call site.
__device__ __forceinline__ float wave_reduce32(float x) {
#if __has_builtin(__builtin_amdgcn_wmma_f32_16x16x4_f32)
    v2f a; a[0] = x;    a[1] = 0.0f;
    v2f b; b[0] = 1.0f; b[1] = 1.0f;
    v8f c = {};
    v8f d = __builtin_amdgcn_wmma_f32_16x16x4_f32(
        /*neg_a=*/false, a, /*neg_b=*/false, b,
        /*c_mod=*/(short)0, c, /*reuse_a=*/false, /*reuse_b=*/false);
    float s = ((d[0] + d[1]) + (d[2] + d[3])) + ((d[4] + d[5]) + (d[6] + d[7]));
    s += __shfl_xor(s, 16, 32);
    return s;
#else
    #pragma unroll
    for (int off = 16; off > 0; off >>= 1) x += __shfl_xor(x, off, 32);
    return x;
#endif
}

__device__ __forceinline__ float diag_iou(const float* t, const float* p) {
    const float invS = 1.0f / (float)S_GRID;
    float txc = t[0] * invS, tyc = t[1] * invS;
    float tx1 = txc - 0.5f * t[2], ty1 = tyc - 0.5f * t[3];
    float tx2 = txc + 0.5f * t[2], ty2 = tyc + 0.5f * t[3];
    float pxc = p[0] * invS, pyc = p[1] * invS;
    float px1 = pxc - 0.5f * p[2], py1 = pyc - 0.5f * p[3];
    float px2 = pxc + 0.5f * p[2], py2 = pyc + 0.5f * p[3];
    float iw = fmaxf(fminf(tx2, px2) - fmaxf(tx1, px1), 0.0f);
    float ih = fmaxf(fminf(ty2, py2) - fmaxf(ty1, py1), 0.0f);
    float inter = iw * ih;
    float at = (tx2 - tx1) * (ty2 - ty1);
    float ap = (px2 - px1) * (py2 - py1);
    return inter / (at + ap - inter);
}

__global__ void __launch_bounds__(256)
yolo_loss_stage1(const float* __restrict__ pred, const float* __restrict__ tgt,
                 float* __restrict__ partials, long long cells) {
    const long long tid = (long long)blockIdx.x * blockDim.x + threadIdx.x;
    const long long stride = (long long)gridDim.x * blockDim.x;

    float acc = 0.0f;
    for (long long c = tid; c < cells; c += stride) {
        float p[CH], t[CH];
        const float2* pp = reinterpret_cast<const float2*>(pred + c * CH);
        const float2* tp = reinterpret_cast<const float2*>(tgt + c * CH);
        #pragma unroll
        for (int k = 0; k < CH / 2; ++k) {
            float2 a = pp[k]; p[2 * k] = a.x; p[2 * k + 1] = a.y;
            float2 b = tp[k]; t[2 * k] = b.x; t[2 * k + 1] = b.y;
        }

        const float objf = ((t[4] + t[9]) > 0.0f) ? 1.0f : 0.0f;
        const float noobjf = 1.0f - objf;

        // no-object confidence loss
        float d4 = p[4] - t[4], d9 = p[9] - t[9];
        float no_obj = noobjf * (d4 * d4 + d9 * d9);

        // class loss (channels 10..29)
        float cls = 0.0f;
        #pragma unroll
        for (int i = 10; i < 30; ++i) {
            float d = p[i] - t[i];
            cls += d * d;
        }
        cls *= objf;

        // per-cell two-box diagonal IoUs (stop-gradient -> just values)
        float iou0 = diag_iou(&t[0], &p[0]);
        float iou1 = diag_iou(&t[5], &p[5]);

        // torch stride-2 responsible-box selection
        float a0 = (iou0 > iou1) ? 1.0f : 0.0f;
        float a1 = (a0 <= iou1) ? 1.0f : 0.0f;
        float w0 = objf * a0, w1 = objf * a1;

        // containing-object confidence loss (target = IoU)
        float c0 = p[4] - iou0, c1 = p[9] - iou1;
        float contain = w0 * c0 * c0 + w1 * c1 * c1;

        // regression loss
        float reg = 0.0f;
        {
            float dx = p[0] - t[0], dy = p[1] - t[1];
            float dsw = sqrtf(p[2] + EPSF) - sqrtf(t[2] + EPSF);
            float dsh = sqrtf(p[3] + EPSF) - sqrtf(t[3] + EPSF);
            reg += w0 * (dx * dx + dy * dy + dsw * dsw + dsh * dsh);
        }
        {
            float dx = p[5] - t[5], dy = p[6] - t[6];
            float dsw = sqrtf(p[7] + EPSF) - sqrtf(t[7] + EPSF);
            float dsh = sqrtf(p[8] + EPSF) - sqrtf(t[8] + EPSF);
            reg += w1 * (dx * dx + dy * dy + dsw * dsw + dsh * dsh);
        }

        acc += 5.0f * reg + contain + 0.5f * no_obj + cls;
    }

    // block reduction: WMMA wave reduce -> LDS across 8 waves
    __shared__ float smem[8];
    float wsum = wave_reduce32(acc);
    const int lane = threadIdx.x & 31;
    const int wid  = threadIdx.x >> 5;
    if (lane == 0) smem[wid] = wsum;
    __syncthreads();
    if (threadIdx.x == 0) {
        float s = 0.0f;
        #pragma unroll
        for (int i = 0; i < 8; ++i) s += smem[i];
        partials[blockIdx.x] = s;
    }
}

__global__ void __launch_bounds__(256)
yolo_loss_stage2(const float* __restrict__ partials, int n, float invN,
                 float* __restrict__ out) {
    float s = 0.0f;
    for (int i = threadIdx.x; i < n; i += 256) s += partials[i];
    __shared__ float smem[8];
    float wsum = wave_reduce32(s);
    const int lane = threadIdx.x & 31;
    const int wid  = threadIdx.x >> 5;
    if (lane == 0) smem[wid] = wsum;
    __syncthreads();
    if (threadIdx.x == 0) {
        float tot = 0.0f;
        #pragma unroll
        for (int i = 0; i < 8; ++i) tot += smem[i];
        out[0] = tot * invN;
    }
}

extern "C" void kernel_launch(void* const* d_in, const int* in_sizes, int n_in,
                              void* d_out, int out_size, void* d_ws, size_t ws_size,
                              hipStream_t stream) {
    const float* pred = (const float*)d_in[0];
    const float* tgt  = (const float*)d_in[1];
    float* out = (float*)d_out;
    float* partials = (float*)d_ws;

    const long long total = (long long)in_sizes[0];      // N*S*S*30
    const long long cells = total / CH;                  // N*S*S
    const int N = (int)(cells / (S_GRID * S_GRID));      // batch size
    const float invN = 1.0f / (float)N;

    int blocks = 1024;
    const int maxb = (int)(ws_size / sizeof(float));
    if (blocks > maxb) blocks = maxb > 0 ? maxb : 1;

    yolo_loss_stage1<<<blocks, 256, 0, stream>>>(pred, tgt, partials, cells);
    yolo_loss_stage2<<<1, 256, 0, stream>>>(partials, blocks, invN, out);
}